// KineticPooling_79439715107413
// MI455X (gfx1250) — compile-verified
//
#include <hip/hip_runtime.h>
#include <hip/hip_bf16.h>

// KineticPooling for MI455X (gfx1250, wave32).
//
// Roofline: 34 MiB traffic (1.5 us @ 23.3 TB/s) vs ~2 GFLOP of sequential 8x8
// LU work => VALU-bound. No GEMM structure exists (batched tiny-matrix
// inversion), so WMMA is intentionally not used; the CDNA5 win here is
// async global->LDS staging of the per-batch MO/d2MO slabs (reused by 64
// configs each) plus wave32-sized scalar elimination with zero divergence.

#define AS_GLOBAL __attribute__((address_space(1)))
#define AS_LOCAL  __attribute__((address_space(3)))

typedef int v4i __attribute__((vector_size(16)));
typedef AS_GLOBAL v4i* gv4i_p;  // global (AS1) 128-bit pointer
typedef AS_LOCAL  v4i* lv4i_p;  // LDS (AS3) 128-bit pointer

#if __has_builtin(__builtin_amdgcn_global_load_async_to_lds_b128) && \
    __has_builtin(__builtin_amdgcn_s_wait_asynccnt)
#define USE_ASYNC_LDS 1
#else
#define USE_ASYNC_LDS 0
#endif

namespace {
constexpr int kBatch         = 8192;
constexpr int kElec          = 16;   // NUP + NDOWN
constexpr int kSpin          = 8;    // NUP == NDOWN == 8
constexpr int kMO            = 32;
constexpr int kConf          = 64;
constexpr int kBatchPerBlock = 4;
constexpr int kThreads       = kBatchPerBlock * kConf;      // 256 = 8 wave32
constexpr int kFloatsPerArr  = kBatchPerBlock * kElec * kMO; // 2048 f32 per array
}  // namespace

__device__ __forceinline__ float fast_rcp(float x) {
  // v_rcp_f32 (~1 ulp) + one Newton step -> ~0.5 ulp, 3 VALU ops instead of a
  // full IEEE divide expansion per pivot.
  float r = __builtin_amdgcn_rcpf(x);
  return r * (2.0f - x * r);
}

// One spin channel for one (batch, config) pair, entirely in this lane's VGPRs:
//   A[i][j] = sMOrow[i*32 + cols[j]]   (8x8, gathered from LDS)
//   B[i][j] = sD2row[i*32 + cols[j]]
//   det  = det(A)       via unpivoted Doolittle LU (inputs are well-conditioned
//                        Gaussian random; pivoting would need indexed register
//                        access and is unnecessary here)
//   tr   = trace(inv(A) @ B)  via per-column forward/back substitution, keeping
//                        only the diagonal element of each solve.
__device__ __forceinline__ void spin_term(const float* __restrict__ sMOrow,
                                          const float* __restrict__ sD2row,
                                          const int cols[8],
                                          float& det, float& tr) {
  float A[8][8];
#pragma unroll
  for (int i = 0; i < 8; ++i) {
#pragma unroll
    for (int j = 0; j < 8; ++j) {
      A[i][j] = sMOrow[i * kMO + cols[j]];
    }
  }

  float rd[8];
  det = 1.0f;
#pragma unroll
  for (int k = 0; k < 8; ++k) {
    const float p = A[k][k];
    det *= p;
    const float r = fast_rcp(p);
    rd[k] = r;
#pragma unroll
    for (int i = k + 1; i < 8; ++i) {
      const float m = A[i][k] * r;
      A[i][k] = m;
#pragma unroll
      for (int j = k + 1; j < 8; ++j) {
        A[i][j] = fmaf(-m, A[k][j], A[i][j]);
      }
    }
  }

  tr = 0.0f;
#pragma unroll
  for (int j = 0; j < 8; ++j) {
    // Solve A x = B[:,j]; accumulate x[j] (diagonal of inv(A)@B).
    float y[8];
#pragma unroll
    for (int i = 0; i < 8; ++i) {
      float v = sD2row[i * kMO + cols[j]];
#pragma unroll
      for (int t = 0; t < i; ++t) v = fmaf(-A[i][t], y[t], v);
      y[i] = v;
    }
    float x[8];
#pragma unroll
    for (int i = 7; i >= j; --i) {  // rows below j are dead and DCE'd
      float v = y[i];
#pragma unroll
      for (int t = i + 1; t < 8; ++t) v = fmaf(-A[i][t], x[t], v);
      x[i] = v * rd[i];
    }
    tr += x[j];
  }
}

__global__ __launch_bounds__(kThreads) void kinetic_pooling_kernel(
    const float* __restrict__ MO, const float* __restrict__ d2MO,
    const int* __restrict__ cup, const int* __restrict__ cdn,
    float* __restrict__ out) {
  __shared__ float sMO[kBatchPerBlock][kElec][kMO];
  __shared__ float sD2[kBatchPerBlock][kElec][kMO];

  const int tid = threadIdx.x;
  const int lb  = tid >> 6;            // local batch 0..3
  const int c   = tid & (kConf - 1);   // config 0..63

  // ---- Stage this block's 4 batches of MO / d2MO into LDS (16 KB). ----
  constexpr int kVec4PerArr = kFloatsPerArr / 4;  // 512 float4 per array
  const float4* gMO = reinterpret_cast<const float4*>(MO) +
                      (size_t)blockIdx.x * kVec4PerArr;
  const float4* gD2 = reinterpret_cast<const float4*>(d2MO) +
                      (size_t)blockIdx.x * kVec4PerArr;
  float4* lMO = reinterpret_cast<float4*>(&sMO[0][0][0]);
  float4* lD2 = reinterpret_cast<float4*>(&sD2[0][0][0]);

#if USE_ASYNC_LDS
#pragma unroll
  for (int k = 0; k < kVec4PerArr / kThreads; ++k) {  // 2 iterations
    const int idx = tid + k * kThreads;
    __builtin_amdgcn_global_load_async_to_lds_b128(
        (gv4i_p)(void*)(gMO + idx), (lv4i_p)(void*)(lMO + idx), 0, 0);
    __builtin_amdgcn_global_load_async_to_lds_b128(
        (gv4i_p)(void*)(gD2 + idx), (lv4i_p)(void*)(lD2 + idx), 0, 0);
  }
  __builtin_amdgcn_s_wait_asynccnt(0);
#else
#pragma unroll
  for (int k = 0; k < kVec4PerArr / kThreads; ++k) {
    const int idx = tid + k * kThreads;
    lMO[idx] = gMO[idx];
    lD2[idx] = gD2[idx];
  }
#endif
  __syncthreads();

  // ---- Per-lane config column indices (contiguous, L2-resident). ----
  int colsU[8], colsD[8];
  const int* __restrict__ pu = cup + c * kSpin;
  const int* __restrict__ pd = cdn + c * kSpin;
#pragma unroll
  for (int j = 0; j < 8; ++j) {
    colsU[j] = pu[j];
    colsD[j] = pd[j];
  }

  // ---- Two spin channels, sequential to bound VGPR pressure. ----
  float detU, trU, detD, trD;
  spin_term(&sMO[lb][0][0],     &sD2[lb][0][0],     colsU, detU, trU);
  spin_term(&sMO[lb][kSpin][0], &sD2[lb][kSpin][0], colsD, detD, trD);

  const int b = blockIdx.x * kBatchPerBlock + lb;
  // out index = blockIdx.x*256 + tid  -> fully coalesced wave32 stores.
  out[(size_t)b * kConf + c] = -0.5f * (trU + trD) * (detU * detD);
}

extern "C" void kernel_launch(void* const* d_in, const int* /*in_sizes*/,
                              int /*n_in*/, void* d_out, int /*out_size*/,
                              void* /*d_ws*/, size_t /*ws_size*/,
                              hipStream_t stream) {
  const float* MO   = (const float*)d_in[0];
  const float* d2MO = (const float*)d_in[1];
  const int*   cup  = (const int*)d_in[2];
  const int*   cdn  = (const int*)d_in[3];
  float*       out  = (float*)d_out;

  const dim3 grid(kBatch / kBatchPerBlock);  // 2048 blocks
  const dim3 block(kThreads);                // 256 threads = 8 wave32
  hipLaunchKernelGGL(kinetic_pooling_kernel, grid, block, 0, stream,
                     MO, d2MO, cup, cdn, out);
}